// VGAEEncoder_80711025427104
// MI455X (gfx1250) — compile-verified
//
#include <hip/hip_runtime.h>
#include <hip/hip_bf16.h>

typedef float v2f __attribute__((ext_vector_type(2)));
typedef float v8f __attribute__((ext_vector_type(8)));

#define C_DIM 32

// ---------------------------------------------------------------------------
// Degree / normalization kernels
// ---------------------------------------------------------------------------
__global__ void k_fill_deg(float* __restrict__ deg, int n) {
    int i = blockIdx.x * blockDim.x + threadIdx.x;
    if (i < n) deg[i] = 1.0f;  // self loop contributes 1
}

__global__ void k_count_deg(const long long* __restrict__ dst,
                            float* __restrict__ deg, int e) {
    int i = blockIdx.x * blockDim.x + threadIdx.x;
    if (i < e) atomicAdd(&deg[(int)dst[i]], 1.0f);
}

__global__ void k_calc_dis(const float* __restrict__ deg,
                           float* __restrict__ dis, int n) {
    int i = blockIdx.x * blockDim.x + threadIdx.x;
    if (i < n) dis[i] = rsqrtf(deg[i]);
}

// ---------------------------------------------------------------------------
// WMMA GEMM: C[n x 32] = op(A)[n x 32] @ W[32 x 32], op = identity or relu.
// One wave produces a 16-row x 32-col strip using V_WMMA_F32_16X16X4_F32.
//
// fp32 A-matrix 16x4 layout: lanes 0-15 hold M=lane {K=k0,k0+1}, lanes 16-31
// hold M=lane-16 {K=k0+2,k0+3}. B (4x16) mirrors this with N striped across
// lanes. C/D 16x16 f32: VGPR v holds M=v (lanes 0-15) / M=v+8 (lanes 16-31),
// N = lane & 15.
// ---------------------------------------------------------------------------
template <bool RELU>
__global__ __launch_bounds__(256) void k_gemm32_wmma(
    const float* __restrict__ A, const float* __restrict__ W,
    float* __restrict__ C, int nrows) {
    int wave = (int)((blockIdx.x * blockDim.x + threadIdx.x) >> 5);
    int lane = threadIdx.x & 31;
    int row0 = wave * 16;
    if (row0 >= nrows) return;  // uniform per wave: EXEC stays all-ones for WMMA

    int m     = lane & 15;
    int khalf = (lane >> 4) << 1;  // 0 for lanes 0-15, 2 for lanes 16-31
    int row   = row0 + m;
    if (row >= nrows) row = nrows - 1;  // clamp (stores are guarded below)
    const float* arow = A + (size_t)row * C_DIM;

    v8f c0 = {};
    v8f c1 = {};
#pragma unroll
    for (int k0 = 0; k0 < C_DIM; k0 += 4) {
        float a0 = arow[k0 + khalf + 0];
        float a1 = arow[k0 + khalf + 1];
        if (RELU) {
            a0 = fmaxf(a0, 0.0f);
            a1 = fmaxf(a1, 0.0f);
        }
        v2f a;
        a.x = a0;
        a.y = a1;

        v2f b0, b1;
        b0.x = W[(k0 + khalf + 0) * C_DIM + m];
        b0.y = W[(k0 + khalf + 1) * C_DIM + m];
        b1.x = W[(k0 + khalf + 0) * C_DIM + 16 + m];
        b1.y = W[(k0 + khalf + 1) * C_DIM + 16 + m];

        c0 = __builtin_amdgcn_wmma_f32_16x16x4_f32(false, a, false, b0,
                                                   (short)0, c0, false, false);
        c1 = __builtin_amdgcn_wmma_f32_16x16x4_f32(false, a, false, b1,
                                                   (short)0, c1, false, false);
    }

    int mhi = (lane >> 4) * 8;
#pragma unroll
    for (int v = 0; v < 8; ++v) {
        int r = row0 + mhi + v;
        if (r < nrows) {
            C[(size_t)r * C_DIM + m]      = c0[v];
            C[(size_t)r * C_DIM + 16 + m] = c1[v];
        }
    }
}

// ---------------------------------------------------------------------------
// agg[i][c] = bias[c] + dis[i]^2 * t[i][c]   (self-loop term + bias; also
// initializes the poisoned destination buffer deterministically)
// ---------------------------------------------------------------------------
__global__ void k_init_agg(const float* __restrict__ t,
                           const float* __restrict__ dis,
                           const float* __restrict__ bias,
                           float* __restrict__ agg, int n) {
    int i = blockIdx.x * blockDim.x + threadIdx.x;
    if (i >= n * C_DIM) return;
    int node = i >> 5;
    int c    = i & (C_DIM - 1);
    float d  = dis[node];
    agg[i]   = bias[c] + d * d * t[i];
}

// ---------------------------------------------------------------------------
// Edge scatter: one wave per edge, lane = channel (coalesced 128B gather of
// t[src] which is L2-resident; f32 atomics into L2-resident agg).
// ---------------------------------------------------------------------------
__global__ __launch_bounds__(256) void k_scatter(
    const long long* __restrict__ src, const long long* __restrict__ dst,
    const float* __restrict__ dis, const float* __restrict__ t,
    float* __restrict__ agg, int nedges) {
    int wave = (int)((blockIdx.x * blockDim.x + threadIdx.x) >> 5);
    int lane = threadIdx.x & 31;
    if (wave >= nedges) return;
    int s = (int)src[wave];
    int d = (int)dst[wave];
    float nrm = dis[s] * dis[d];
    float val = t[(size_t)s * C_DIM + lane] * nrm;
    atomicAdd(&agg[(size_t)d * C_DIM + lane], val);
}

// ---------------------------------------------------------------------------
extern "C" void kernel_launch(void* const* d_in, const int* in_sizes, int n_in,
                              void* d_out, int out_size, void* d_ws,
                              size_t ws_size, hipStream_t stream) {
    const float*     x   = (const float*)d_in[0];
    const long long* ei  = (const long long*)d_in[1];
    const float*     W1  = (const float*)d_in[2];
    const float*     b1  = (const float*)d_in[3];
    const float*     W2  = (const float*)d_in[4];
    const float*     b2  = (const float*)d_in[5];
    const float*     Wmu = (const float*)d_in[6];
    const float*     bmu = (const float*)d_in[7];
    const float*     Wlv = (const float*)d_in[8];
    const float*     blv = (const float*)d_in[9];

    const int n = in_sizes[0] / C_DIM;   // 100000
    const int e = in_sizes[1] / 2;       // 1600000
    const long long* src = ei;
    const long long* dst = ei + e;

    float* ws   = (float*)d_ws;
    float* deg  = ws;                 // n
    float* dis  = deg + n;            // n
    float* t    = dis + n;            // n*32
    float* aggA = t + (size_t)n * C_DIM;     // n*32
    float* aggB = aggA + (size_t)n * C_DIM;  // n*32

    float* mu = (float*)d_out;
    float* lv = mu + (size_t)n * C_DIM;

    const int TPB       = 256;
    const int blkN      = (n + TPB - 1) / TPB;
    const int blkE      = (e + TPB - 1) / TPB;
    const int blkNC     = (n * C_DIM + TPB - 1) / TPB;
    const int wavesGemm = (n + 15) / 16;
    const int blkGemm   = (wavesGemm + 7) / 8;   // 8 waves per 256-thread block
    const int blkScat   = (e + 7) / 8;           // 1 wave per edge

    // Normalization: deg = 1 + in-degree; dis = deg^-1/2
    k_fill_deg<<<blkN, TPB, 0, stream>>>(deg, n);
    k_count_deg<<<blkE, TPB, 0, stream>>>(dst, deg, e);
    k_calc_dis<<<blkN, TPB, 0, stream>>>(deg, dis, n);

    // Layer 1: t = x @ W1 ; aggA = b1 + dis^2*t + scatter
    k_gemm32_wmma<false><<<blkGemm, TPB, 0, stream>>>(x, W1, t, n);
    k_init_agg<<<blkNC, TPB, 0, stream>>>(t, dis, b1, aggA, n);
    k_scatter<<<blkScat, TPB, 0, stream>>>(src, dst, dis, t, aggA, e);

    // Layer 2: t = relu(aggA) @ W2 ; aggB = b2 + dis^2*t + scatter
    k_gemm32_wmma<true><<<blkGemm, TPB, 0, stream>>>(aggA, W2, t, n);
    k_init_agg<<<blkNC, TPB, 0, stream>>>(t, dis, b2, aggB, n);
    k_scatter<<<blkScat, TPB, 0, stream>>>(src, dst, dis, t, aggB, e);

    // mu head: t = relu(aggB) @ Wmu ; mu = bmu + dis^2*t + scatter
    k_gemm32_wmma<true><<<blkGemm, TPB, 0, stream>>>(aggB, Wmu, t, n);
    k_init_agg<<<blkNC, TPB, 0, stream>>>(t, dis, bmu, mu, n);
    k_scatter<<<blkScat, TPB, 0, stream>>>(src, dst, dis, t, mu, e);

    // logvar head: t = relu(aggB) @ Wlv ; lv = blv + dis^2*t + scatter
    k_gemm32_wmma<true><<<blkGemm, TPB, 0, stream>>>(aggB, Wlv, t, n);
    k_init_agg<<<blkNC, TPB, 0, stream>>>(t, dis, blv, lv, n);
    k_scatter<<<blkScat, TPB, 0, stream>>>(src, dst, dis, t, lv, e);
}